// NormalizeBoardToPerspectiveLayer_40785009442996
// MI455X (gfx1250) — compile-verified
//
#include <hip/hip_runtime.h>
#include <stdint.h>

// Board normalization + per-sample rot90 for MI455X (gfx1250, wave32).
// Memory-bound: ~420 MB total traffic -> ~18 us floor at 23.3 TB/s.
// CDNA5 path: GLOBAL_LOAD_ASYNC_TO_LDS_B128 (ASYNCcnt DMA) stages boards in LDS;
// rotation is a branchless affine gather from LDS; stores are b128 coalesced.

#define THREADS          256
#define BOARDS_PER_BLOCK 8
#define CELLS            400                                  // 20*20
#define BYTES_PER_BOARD  1600                                 // 400 * 4
#define VEC_PER_BLOCK    (BOARDS_PER_BLOCK * BYTES_PER_BOARD / 16)  // 800 x 16B in
#define VEC4_OUT         (BOARDS_PER_BLOCK * CELLS / 4)             // 800 x float4 out

__device__ __forceinline__ float norm_cell(float v, int p) {
    int iv = (int)v;  // values are exact integers in {-1..3}
    return (iv < 0) ? -1.0f : (float)((iv + 4 - p) & 3);
}

__global__ __launch_bounds__(THREADS)
void normalize_rot_kernel(const float* __restrict__ board,
                          const int*   __restrict__ pid,
                          float*       __restrict__ out,
                          int nBoards)
{
    __shared__ __align__(16) float smem[BOARDS_PER_BLOCK * CELLS];
    __shared__ int4 sM[BOARDS_PER_BLOCK];   // (a, b, c, pid) per board: src = a*i + b*j + c

    const int tid       = threadIdx.x;
    const int boardBase = blockIdx.x * BOARDS_PER_BLOCK;
    int boardsHere = nBoards - boardBase;
    if (boardsHere <= 0) return;
    if (boardsHere > BOARDS_PER_BLOCK) boardsHere = BOARDS_PER_BLOCK;

    // ---- Stage A: async DMA of raw boards (contiguous) into LDS, 16B per lane ----
    const int totalVec = (boardsHere * BYTES_PER_BOARD) >> 4;
    const uint64_t gbase   = (uint64_t)(uintptr_t)(board + (size_t)boardBase * CELLS);
    const unsigned ldsBase = (unsigned)(uintptr_t)(&smem[0]);  // low 32 bits = LDS byte addr

    #pragma unroll
    for (int it = 0; it < (VEC_PER_BLOCK + THREADS - 1) / THREADS; ++it) {
        int t = tid + it * THREADS;
        if (t < totalVec) {
            uint64_t gaddr = gbase + ((uint64_t)t << 4);
            unsigned laddr = ldsBase + ((unsigned)t << 4);
            asm volatile("global_load_async_to_lds_b128 %0, %1, off"
                         :: "v"(laddr), "v"(gaddr)
                         : "memory");
        }
    }
    asm volatile("s_wait_asynccnt 0x0" ::: "memory");
    __syncthreads();

    // ---- Stage B: per-board rotation as affine map from RAW corners ----
    // normalized corner == 0  <=>  raw corner == pid  (corners in {-1,0..3}, pid in 0..3)
    if (tid < boardsHere) {
        int p = pid[boardBase + tid];
        const float fp = (float)p;
        const float* bb = &smem[tid * CELLS];
        float c0 = bb[0];    // TL  b[0,0]
        float c1 = bb[19];   // TR  b[0,19]
        float c2 = bb[399];  // BR  b[19,19]
        float c3 = bb[380];  // BL  b[19,0]
        // k=0 default (argmax of all-zeros -> 0): src = 20i + j
        int a = 20, b = 1, c = 0;
        if      (c0 == fp) { /* k=0 */ }
        else if (c1 == fp) { a = -1;  b = 20;  c = 19;  }  // k=1: b[j,19-i]
        else if (c2 == fp) { a = -20; b = -1;  c = 399; }  // k=2: b[19-i,19-j]
        else if (c3 == fp) { a = 1;   b = -20; c = 380; }  // k=3: b[19-j,i]
        sM[tid] = make_int4(a, b, c, p);
    }
    __syncthreads();

    // ---- Stage C: affine gather + normalize, b128 coalesced stores ----
    float* obase = out + (size_t)boardBase * CELLS;

    if (boardsHere == BOARDS_PER_BLOCK) {
        // Full block: 800 float4 stores; 20 % 4 == 0 so each quad stays in one row.
        #pragma unroll
        for (int it = 0; it < (VEC4_OUT + THREADS - 1) / THREADS; ++it) {
            int t = tid + it * THREADS;
            if (t < VEC4_OUT) {
                int bd = t / 100;              // 100 quads per board
                int r  = t - bd * 100;
                int q  = r * 4;                // flat index in board, multiple of 4
                int i  = q / 20;
                int j  = q - i * 20;           // j in {0,4,8,12,16}
                int4 m = sM[bd];
                int src = m.x * i + m.y * j + m.z;
                const float* bb = &smem[bd * CELLS];
                float v0 = bb[src];
                float v1 = bb[src + m.y];
                float v2 = bb[src + 2 * m.y];
                float v3 = bb[src + 3 * m.y];
                float4 o;
                o.x = norm_cell(v0, m.w);
                o.y = norm_cell(v1, m.w);
                o.z = norm_cell(v2, m.w);
                o.w = norm_cell(v3, m.w);
                *(float4*)(obase + (size_t)t * 4) = o;   // 16B aligned
            }
        }
    } else {
        // Ragged tail (never hit for B % 8 == 0): scalar generic path.
        const int totalOut = boardsHere * CELLS;
        for (int o = tid; o < totalOut; o += THREADS) {
            int bd = o / CELLS;
            int q  = o - bd * CELLS;
            int i  = q / 20;
            int j  = q - i * 20;
            int4 m = sM[bd];
            int src = m.x * i + m.y * j + m.z;
            obase[o] = norm_cell(smem[bd * CELLS + src], m.w);
        }
    }
}

extern "C" void kernel_launch(void* const* d_in, const int* in_sizes, int n_in,
                              void* d_out, int out_size, void* d_ws, size_t ws_size,
                              hipStream_t stream)
{
    const float* board = (const float*)d_in[0];  // [B,20,20] float32
    const int*   pid   = (const int*)d_in[1];    // [B] int32
    float*       out   = (float*)d_out;          // [B,20,20] float32

    const int nBoards = (n_in > 1) ? in_sizes[1] : (in_sizes[0] / CELLS);
    const int grid    = (nBoards + BOARDS_PER_BLOCK - 1) / BOARDS_PER_BLOCK;

    hipLaunchKernelGGL(normalize_rot_kernel, dim3(grid), dim3(THREADS), 0, stream,
                       board, pid, out, nBoards);
}